// ModelNew_69879117906329
// MI455X (gfx1250) — compile-verified
//
#include <hip/hip_runtime.h>

// ---------------------------------------------------------------------------
// 2-layer GRU for MI455X (gfx1250).
//   fp32 WMMA (V_WMMA_F32_16X16X4_F32) + Tensor Data Mover (TDM) staging
//   of all GEMM operands into LDS (double-buffered, TENSORcnt-tracked).
//   T=512, B=64, I=H=1024, G=3H=3072, L=2
// ---------------------------------------------------------------------------

typedef float v2f __attribute__((ext_vector_type(2)));
typedef float v8f __attribute__((ext_vector_type(8)));
typedef unsigned int v4u __attribute__((ext_vector_type(4)));
typedef int v8i __attribute__((ext_vector_type(8)));
typedef int v4i __attribute__((ext_vector_type(4)));

#define HID   1024
#define G3    3072
#define BATCH 64
#define TSTEP 512
#define KCH   32            // K-chunk staged per TDM round
#define LDSP  33            // padded LDS row stride (TDM pad: +1 dword / 32)

static __device__ __forceinline__ v8f wmma4(v2f a, v2f b, v8f c) {
  // D = A(16x4 f32) * B(4x16 f32) + C(16x16 f32)
  return __builtin_amdgcn_wmma_f32_16x16x4_f32(
      false, a, false, b, (short)0, c, false, false);
}

static __device__ __forceinline__ float sigmoidf_fast(float x) {
  return 1.0f / (1.0f + __expf(-x));
}

// ---------------------------------------------------------------------------
// TDM: DMA a 2-D fp32 tile (tile_rows x tile_cols) from global to LDS.
// LDS destination gets +1 dword padding every 32 dwords (row stride = 33
// dwords for tile_cols == 32) to avoid bank conflicts on column reads.
// Descriptor layout per CDNA5 ISA (D# groups 0/1; 2-D).
// ---------------------------------------------------------------------------
static __device__ __forceinline__ void tdm_load_tile_f32(
    unsigned lds_off_bytes, const float* gsrc,
    unsigned tile_rows, unsigned tile_cols,
    unsigned row_stride_elems, unsigned tensor_cols, unsigned tensor_rows) {
  const unsigned long long ga = (unsigned long long)(uintptr_t)gsrc;
  v4u g0;
  g0.x = 0x1u;                                   // count=1, user descriptor
  g0.y = lds_off_bytes;                          // lds_addr
  g0.z = (unsigned)ga;                           // global_addr[31:0]
  g0.w = (unsigned)(ga >> 32) | (2u << 30);      // global_addr[56:32] | type=2
  v8i g1;
  // d0: data_size=2(4B)<<16 | pad_enable<<20 | pad_interval=4(32dw)<<22
  //     | pad_amount=0(1dw)<<25 ; workgroup_mask=0
  g1[0] = (int)((2u << 16) | (1u << 20) | (4u << 22));
  g1[1] = (int)((tensor_cols & 0xFFFFu) << 16);                    // dim0 lo16
  g1[2] = (int)((tensor_cols >> 16) | ((tensor_rows & 0xFFFFu) << 16));
  g1[3] = (int)((tensor_rows >> 16) | (tile_cols << 16));          // tile_dim0
  g1[4] = (int)(tile_rows & 0xFFFFu);            // tile_dim1 (tile_dim2=0)
  g1[5] = (int)row_stride_elems;                 // tensor_dim0_stride[31:0]
  g1[6] = 0;                                     // stride hi / dim1_stride lo
  g1[7] = 0;
  const v4i z4 = {0, 0, 0, 0};                   // 2-D: groups 2/3 unused
  const v8i z8 = {0, 0, 0, 0, 0, 0, 0, 0};
  __builtin_amdgcn_tensor_load_to_lds(g0, g1, z4, z4, z8, 0);
}

static __device__ __forceinline__ unsigned lds_off(const void* p) {
  return (unsigned)(uintptr_t)p;                 // low 32 bits = LDS byte offset
}

// ---------------------------------------------------------------------------
// Phase 1: xg[m,g] = in[m,:] . w[g,:] + bias[g]
// Grid: (M/64, G3/64), block 256 (8 waves: 4 m-groups x 2 n-groups).
// Each chunk: TDM stages A(64xKCH) and B(64xKCH) tiles; each wave does a
// 16x32 output tile (two 16x16 f32 accumulators sharing the A fragment).
// ---------------------------------------------------------------------------
__global__ void __launch_bounds__(256) gru_xgates_gemm(
    const float* __restrict__ in,    // M x 1024
    const float* __restrict__ w,     // 3072 x 1024
    const float* __restrict__ bias,  // 3072
    float* __restrict__ xg)          // M x 3072
{
  __shared__ float sA[2][64][LDSP];
  __shared__ float sB[2][64][LDSP];

  const int lane = threadIdx.x & 31;
  const int wave = threadIdx.x >> 5;
  const int mg = wave & 3;             // m-group within WG
  const int ng = wave >> 2;            // n-group within WG
  const int lm = lane & 15;
  const int kh = (lane >> 4) << 1;     // K base within chunk-of-4: 0 or 2
  const int mblk = blockIdx.x * 64;
  const int nblk = blockIdx.y * 64;

  const float* abase = in + (size_t)mblk * HID;
  const float* bbase = w + (size_t)nblk * HID;

  v8f acc0 = {};
  v8f acc1 = {};

  if (wave == 0) {
    tdm_load_tile_f32(lds_off(&sA[0][0][0]), abase, 64, KCH, HID, HID, 64);
    tdm_load_tile_f32(lds_off(&sB[0][0][0]), bbase, 64, KCH, HID, HID, 64);
  }

  for (int c = 0; c < HID / KCH; ++c) {
    const int buf = c & 1;
    if (wave == 0) {
      if (c + 1 < HID / KCH) {
        const int kn = (c + 1) * KCH;
        tdm_load_tile_f32(lds_off(&sA[buf ^ 1][0][0]), abase + kn, 64, KCH, HID, HID, 64);
        tdm_load_tile_f32(lds_off(&sB[buf ^ 1][0][0]), bbase + kn, 64, KCH, HID, HID, 64);
        __builtin_amdgcn_s_wait_tensorcnt(2);   // chunk c complete
      } else {
        __builtin_amdgcn_s_wait_tensorcnt(0);
      }
    }
    __syncthreads();                            // publish chunk c
#pragma unroll
    for (int k4 = 0; k4 < KCH; k4 += 4) {
      const int k = k4 + kh;
      v2f a, b0, b1;
      a.x  = sA[buf][mg * 16 + lm][k];
      a.y  = sA[buf][mg * 16 + lm][k + 1];
      b0.x = sB[buf][ng * 32 + lm][k];
      b0.y = sB[buf][ng * 32 + lm][k + 1];
      b1.x = sB[buf][ng * 32 + 16 + lm][k];
      b1.y = sB[buf][ng * 32 + 16 + lm][k + 1];
      acc0 = wmma4(a, b0, acc0);
      acc1 = wmma4(a, b1, acc1);
    }
    __syncthreads();                            // chunk c consumed
  }

  const int   n0   = nblk + ng * 32;
  const int   mrow = mblk + mg * 16 + ((lane >> 4) << 3);
  const float bia0 = bias[n0 + lm];
  const float bia1 = bias[n0 + 16 + lm];
#pragma unroll
  for (int r = 0; r < 8; ++r) {
    xg[(size_t)(mrow + r) * G3 + n0 + lm]      = acc0[r] + bia0;
    xg[(size_t)(mrow + r) * G3 + n0 + 16 + lm] = acc1[r] + bia1;
  }
}

// ---------------------------------------------------------------------------
// Phase 2: one sequential GRU step.
// Grid: 32 blocks x 256 thr; WG covers 64(batch) x 32(hidden cols).
// Per chunk TDM stages: h tile (64xKCH) + the three w_hh gate blocks
// (3 x 32 x KCH). Each wave: 16(b) x 16(j) tile of all three gates, then
// fused sigmoid/tanh gate math + h update.
// ---------------------------------------------------------------------------
__global__ void __launch_bounds__(256) gru_step(
    const float* __restrict__ h,       // 64 x 1024
    const float* __restrict__ whh,     // 3072 x 1024
    const float* __restrict__ bhh,     // 3072
    const float* __restrict__ xg_t,    // 64 x 3072
    float* __restrict__ h_new,         // 64 x 1024
    float* __restrict__ seq_out)       // 64 x 1024 or nullptr
{
  __shared__ float sH[2][64][LDSP];
  __shared__ float sW[2][96][LDSP];    // rows 0-31:r, 32-63:z, 64-95:n

  const int lane = threadIdx.x & 31;
  const int wave = threadIdx.x >> 5;
  const int bg = wave & 3;             // batch group (0..3)
  const int jg = wave >> 2;            // col group (0..1)
  const int lm = lane & 15;
  const int kh = (lane >> 4) << 1;
  const int jblk = blockIdx.x * 32;

  const float* wr = whh + (size_t)jblk * HID;
  const float* wz = whh + (size_t)(HID + jblk) * HID;
  const float* wn = whh + (size_t)(2 * HID + jblk) * HID;

  v8f acc_r = {};
  v8f acc_z = {};
  v8f acc_n = {};

  if (wave == 0) {
    tdm_load_tile_f32(lds_off(&sH[0][0][0]),  h,  64, KCH, HID, HID, 64);
    tdm_load_tile_f32(lds_off(&sW[0][0][0]),  wr, 32, KCH, HID, HID, 32);
    tdm_load_tile_f32(lds_off(&sW[0][32][0]), wz, 32, KCH, HID, HID, 32);
    tdm_load_tile_f32(lds_off(&sW[0][64][0]), wn, 32, KCH, HID, HID, 32);
  }

  for (int c = 0; c < HID / KCH; ++c) {
    const int buf = c & 1;
    if (wave == 0) {
      if (c + 1 < HID / KCH) {
        const int kn = (c + 1) * KCH;
        tdm_load_tile_f32(lds_off(&sH[buf ^ 1][0][0]),  h + kn,  64, KCH, HID, HID, 64);
        tdm_load_tile_f32(lds_off(&sW[buf ^ 1][0][0]),  wr + kn, 32, KCH, HID, HID, 32);
        tdm_load_tile_f32(lds_off(&sW[buf ^ 1][32][0]), wz + kn, 32, KCH, HID, HID, 32);
        tdm_load_tile_f32(lds_off(&sW[buf ^ 1][64][0]), wn + kn, 32, KCH, HID, HID, 32);
        __builtin_amdgcn_s_wait_tensorcnt(4);
      } else {
        __builtin_amdgcn_s_wait_tensorcnt(0);
      }
    }
    __syncthreads();
#pragma unroll
    for (int k4 = 0; k4 < KCH; k4 += 4) {
      const int k = k4 + kh;
      v2f a, br, bz, bn;
      a.x  = sH[buf][bg * 16 + lm][k];
      a.y  = sH[buf][bg * 16 + lm][k + 1];
      br.x = sW[buf][jg * 16 + lm][k];
      br.y = sW[buf][jg * 16 + lm][k + 1];
      bz.x = sW[buf][32 + jg * 16 + lm][k];
      bz.y = sW[buf][32 + jg * 16 + lm][k + 1];
      bn.x = sW[buf][64 + jg * 16 + lm][k];
      bn.y = sW[buf][64 + jg * 16 + lm][k + 1];
      acc_r = wmma4(a, br, acc_r);
      acc_z = wmma4(a, bz, acc_z);
      acc_n = wmma4(a, bn, acc_n);
    }
    __syncthreads();
  }

  const int   j    = jblk + jg * 16 + lm;
  const float br_b = bhh[j];
  const float bz_b = bhh[HID + j];
  const float bn_b = bhh[2 * HID + j];
  const int   brow = bg * 16 + ((lane >> 4) << 3);

#pragma unroll
  for (int r = 0; r < 8; ++r) {
    const int b = brow + r;
    const float hr = acc_r[r] + br_b;
    const float hz = acc_z[r] + bz_b;
    const float hn = acc_n[r] + bn_b;
    const float xr = xg_t[(size_t)b * G3 + j];
    const float xz = xg_t[(size_t)b * G3 + HID + j];
    const float xn = xg_t[(size_t)b * G3 + 2 * HID + j];
    const float rg = sigmoidf_fast(xr + hr);
    const float zg = sigmoidf_fast(xz + hz);
    const float ng = tanhf(xn + rg * hn);
    const float hp = h[(size_t)b * HID + j];
    const float ho = (1.0f - zg) * ng + zg * hp;
    h_new[(size_t)b * HID + j] = ho;
    if (seq_out) seq_out[(size_t)b * HID + j] = ho;
  }
}

// ---------------------------------------------------------------------------
// Host orchestration (graph-capture safe: stream launches + async D2D only).
// d_in order: x, h0, w_ih, w_hh, b_ih, b_hh
// d_ws layout: xg (512*64*3072) | seq0 (512*64*1024) | hA (64*1024) | hB
// ---------------------------------------------------------------------------
extern "C" void kernel_launch(void* const* d_in, const int* in_sizes, int n_in,
                              void* d_out, int out_size, void* d_ws, size_t ws_size,
                              hipStream_t stream) {
  (void)in_sizes; (void)n_in; (void)out_size; (void)ws_size;

  const float* x    = (const float*)d_in[0];
  const float* h0   = (const float*)d_in[1];
  const float* w_ih = (const float*)d_in[2];
  const float* w_hh = (const float*)d_in[3];
  const float* b_ih = (const float*)d_in[4];
  const float* b_hh = (const float*)d_in[5];
  float* out = (float*)d_out;

  float* xg   = (float*)d_ws;                               // 512*64*3072
  float* seq0 = xg + (size_t)TSTEP * BATCH * G3;            // 512*64*1024
  float* hA   = seq0 + (size_t)TSTEP * BATCH * HID;         // 64*1024
  float* hB   = hA + (size_t)BATCH * HID;                   // 64*1024

  const size_t M = (size_t)TSTEP * BATCH;                   // 32768

  for (int l = 0; l < 2; ++l) {
    const float* in_l  = (l == 0) ? x : seq0;
    const float* wih_l = w_ih + (size_t)l * G3 * HID;
    const float* whh_l = w_hh + (size_t)l * G3 * HID;
    const float* bih_l = b_ih + (size_t)l * G3;
    const float* bhh_l = b_hh + (size_t)l * G3;

    dim3 g1((unsigned)(M / 64), G3 / 64);
    gru_xgates_gemm<<<g1, 256, 0, stream>>>(in_l, wih_l, bih_l, xg);

    (void)hipMemcpyAsync(hA, h0 + (size_t)l * BATCH * HID,
                         (size_t)BATCH * HID * sizeof(float),
                         hipMemcpyDeviceToDevice, stream);

    float* hc = hA;
    float* hn = hB;
    for (int t = 0; t < TSTEP; ++t) {
      float* so = (l == 0) ? (seq0 + (size_t)t * BATCH * HID) : nullptr;
      gru_step<<<32, 256, 0, stream>>>(
          hc, whh_l, bhh_l, xg + (size_t)t * BATCH * G3, hn, so);
      float* tmp = hc; hc = hn; hn = tmp;
    }

    (void)hipMemcpyAsync(out + (size_t)l * BATCH * HID, hc,
                         (size_t)BATCH * HID * sizeof(float),
                         hipMemcpyDeviceToDevice, stream);
  }
}